// PDNetPostProcessor_51402168598855
// MI455X (gfx1250) — compile-verified
//
#include <hip/hip_runtime.h>
#include <hip/hip_bf16.h>

// ---------------- problem constants ----------------
#define NB       16      // batch
#define NCLS     80      // classes
#define NCAND    3267    // 1000 + 1000 + 950 + 247 + 70
#define MSEL     1000    // M_CAND / PRE_NMS_TOP_N
#define THR_LOGIT (-2.9444389791664403f)   // log(0.05/0.95): sigmoid(x)>0.05 <=> x>this
#define NEGF     (-1e30f)
#define NMS_T    0.6f

// level geometry
// hw:      15200, 3800, 950, 247, 70   (sum 20267)
// strides: 8, 16, 32, 64, 128
// cand cnt:1000, 1000, 950, 247, 70 ; bases 0,1000,2000,2950,3197

// ---------------- workspace layout (bytes) ----------------
constexpr size_t WS_ROWMAX = 0;                                   // NB*19008 f32
constexpr size_t WS_CANDROW = WS_ROWMAX + (size_t)NB*19008*4;     // NB*3280 i32
constexpr size_t WS_CANDLVL = WS_CANDROW + (size_t)NB*3280*4;     // NB*3280 i32
constexpr size_t WS_CANDLOG = WS_CANDLVL + (size_t)NB*3280*4;     // NB*261376 f32
constexpr size_t WS_SELSC   = WS_CANDLOG + (size_t)NB*261376*4;   // NB*1024 f32
constexpr size_t WS_SELIDX  = WS_SELSC + (size_t)NB*1024*4;       // NB*1024 i32
constexpr size_t WS_SELBOX  = WS_SELIDX + (size_t)NB*1024*4;      // NB*1024*4 f32
constexpr size_t WS_NBBOX   = WS_SELBOX + (size_t)NB*4096*4;      // NB*1024*4 f32
constexpr size_t WS_SELLAB  = WS_NBBOX + (size_t)NB*4096*4;       // NB*1024 i32
constexpr size_t WS_MASK    = WS_SELLAB + (size_t)NB*1024*4;      // NB*1024*32 u32
constexpr size_t WS_KEEP    = WS_MASK + (size_t)NB*32768*4;       // NB*32 u32
constexpr size_t WS_CNT     = WS_KEEP + (size_t)NB*32*4;          // 256 i32 counters

// ---------------- vector types ----------------
typedef float v2f  __attribute__((ext_vector_type(2)));
typedef float v8f  __attribute__((ext_vector_type(8)));
typedef unsigned int u32x4 __attribute__((ext_vector_type(4)));
typedef int i32x8 __attribute__((ext_vector_type(8)));
typedef int i32x4 __attribute__((ext_vector_type(4)));

__device__ __forceinline__ unsigned f2u(float f) {
    unsigned u = __float_as_uint(f);
    return (u & 0x80000000u) ? ~u : (u | 0x80000000u);
}

// ---------------- K0: zero atomic counters ----------------
__global__ void k_init(int* cnt) {
    cnt[threadIdx.x] = 0;
}

// ---------------- K1: per-row max logit over 80 classes (levels 0,1) ----------------
__global__ void __launch_bounds__(256) k_rowmax(const float* __restrict__ cls0,
                                                const float* __restrict__ cls1,
                                                float* __restrict__ rowmax) {
    int t = blockIdx.x * blockDim.x + threadIdx.x;
    if (t >= NB * 19000) return;
    int img = t / 19000, r = t - img * 19000;
    const float* src; int row, hw;
    if (r < 15200) { src = cls0; row = r; hw = 15200; }
    else           { src = cls1; row = r - 15200; hw = 3800; }
    const float* p = src + (size_t)img * NCLS * hw + row;
    float m = -3.4e38f;
    #pragma unroll 4
    for (int c = 0; c < NCLS; ++c) m = fmaxf(m, p[(size_t)c * hw]);
    rowmax[img * 19008 + r] = m;
}

// ---------------- K2: per-(image,level) top-1000 row selection (radix select) ------
__global__ void __launch_bounds__(1024) k_level_select(const float* __restrict__ rowmax,
                                                       int* __restrict__ candRow,
                                                       int* __restrict__ candLvl,
                                                       int* __restrict__ cnt) {
    int img = blockIdx.x >> 1, level = blockIdx.x & 1;
    int n = level ? 3800 : 15200;
    const float* v = rowmax + img * 19008 + (level ? 15200 : 0);
    __shared__ unsigned bins[16];
    __shared__ unsigned s_prefix;
    __shared__ int s_k;
    if (threadIdx.x == 0) { s_prefix = 0u; s_k = MSEL; }
    __syncthreads();
    for (int pass = 7; pass >= 0; --pass) {
        if (threadIdx.x < 16) bins[threadIdx.x] = 0u;
        __syncthreads();
        unsigned pfx = s_prefix;
        int shift = pass * 4;
        for (int i = threadIdx.x; i < n; i += blockDim.x) {
            unsigned u = f2u(v[i]);
            bool match = (pass == 7) || ((u >> (shift + 4)) == pfx);
            if (match) atomicAdd(&bins[(u >> shift) & 15u], 1u);
        }
        __syncthreads();
        if (threadIdx.x == 0) {
            int k = s_k; unsigned cum = 0; int chosen = 0;
            for (int b = 15; b >= 0; --b) {
                if ((int)(cum + bins[b]) >= k) { chosen = b; s_k = k - (int)cum; break; }
                cum += bins[b];
            }
            s_prefix = (s_prefix << 4) | (unsigned)chosen;
        }
        __syncthreads();
    }
    unsigned T = s_prefix;
    int* counter = cnt + img * 2 + level;
    int* orow = candRow + img * 3280 + level * 1000;
    int* olvl = candLvl + img * 3280 + level * 1000;
    for (int i = threadIdx.x; i < n; i += blockDim.x) {
        if (f2u(v[i]) >= T) {
            int pos = atomicAdd(counter, 1);
            if (pos < MSEL) { orow[pos] = i; olvl[pos] = level; }
        }
    }
}

// ---------------- K2b: identity candidates for levels 2..4 ----------------
__global__ void __launch_bounds__(256) k_fill_tail(int* __restrict__ candRow,
                                                   int* __restrict__ candLvl) {
    int t = blockIdx.x * blockDim.x + threadIdx.x;
    if (t >= NB * 1267) return;
    int img = t / 1267, j = t - img * 1267;
    int level, row;
    if (j < 950)       { level = 2; row = j; }
    else if (j < 1197) { level = 3; row = j - 950; }
    else               { level = 4; row = j - 1197; }
    candRow[img * 3280 + 2000 + j] = row;
    candLvl[img * 3280 + 2000 + j] = level;
}

// ---------------- K3: gather candidate logits (3267x80 per image) ----------------
__global__ void __launch_bounds__(256) k_gather(const float* __restrict__ c0, const float* __restrict__ c1,
                                                const float* __restrict__ c2, const float* __restrict__ c3,
                                                const float* __restrict__ c4,
                                                const int* __restrict__ candRow, const int* __restrict__ candLvl,
                                                float* __restrict__ clog) {
    int t = blockIdx.x * blockDim.x + threadIdx.x;
    if (t >= NB * NCAND * NCLS) return;
    int img = t / (NCAND * NCLS);
    int rem = t - img * (NCAND * NCLS);
    int cand = rem / NCLS, c = rem - cand * NCLS;
    int lvl = candLvl[img * 3280 + cand];
    int row = candRow[img * 3280 + cand];
    const float* base; int hw;
    if (lvl == 0)      { base = c0; hw = 15200; }
    else if (lvl == 1) { base = c1; hw = 3800; }
    else if (lvl == 2) { base = c2; hw = 950; }
    else if (lvl == 3) { base = c3; hw = 247; }
    else               { base = c4; hw = 70; }
    clog[(size_t)img * 261376 + rem] = base[(size_t)img * NCLS * hw + (size_t)c * hw + row];
}

// ---------------- K4: per-image flat top-1000 over 261360 logits ----------------
__global__ void __launch_bounds__(1024) k_flat_select(const float* __restrict__ clog,
                                                      float* __restrict__ selScore,
                                                      int* __restrict__ selIdx,
                                                      int* __restrict__ cnt) {
    int img = blockIdx.x;
    const float* v = clog + (size_t)img * 261376;
    const int n = NCAND * NCLS;
    __shared__ unsigned bins[16];
    __shared__ unsigned s_prefix;
    __shared__ int s_k;
    if (threadIdx.x == 0) { s_prefix = 0u; s_k = MSEL; }
    __syncthreads();
    for (int pass = 7; pass >= 0; --pass) {
        if (threadIdx.x < 16) bins[threadIdx.x] = 0u;
        __syncthreads();
        unsigned pfx = s_prefix;
        int shift = pass * 4;
        for (int i = threadIdx.x; i < n; i += blockDim.x) {
            float f = v[i];
            if (!(f > THR_LOGIT)) f = NEGF;   // pre-NMS threshold in logit domain
            unsigned u = f2u(f);
            bool match = (pass == 7) || ((u >> (shift + 4)) == pfx);
            if (match) atomicAdd(&bins[(u >> shift) & 15u], 1u);
        }
        __syncthreads();
        if (threadIdx.x == 0) {
            int k = s_k; unsigned cum = 0; int chosen = 0;
            for (int b = 15; b >= 0; --b) {
                if ((int)(cum + bins[b]) >= k) { chosen = b; s_k = k - (int)cum; break; }
                cum += bins[b];
            }
            s_prefix = (s_prefix << 4) | (unsigned)chosen;
        }
        __syncthreads();
    }
    unsigned T = s_prefix;
    for (int i = threadIdx.x; i < n; i += blockDim.x) {
        float f = v[i];
        if (!(f > THR_LOGIT)) f = NEGF;
        if (f2u(f) >= T) {
            int pos = atomicAdd(cnt + img, 1);
            if (pos < MSEL) {
                selScore[img * 1024 + pos] = f;
                selIdx[img * 1024 + pos] = i;
            }
        }
    }
}

// ---------------- K5: per-image bitonic sort of 1024 (desc by score) ----------------
__global__ void __launch_bounds__(1024) k_sort(float* __restrict__ selScore, int* __restrict__ selIdx) {
    __shared__ float key[1024];
    __shared__ int pay[1024];
    int t = threadIdx.x, img = blockIdx.x;
    float v = (t < MSEL) ? selScore[img * 1024 + t] : NEGF;
    key[t] = -v;                               // ascending sort of -score == descending score
    pay[t] = (t < MSEL) ? selIdx[img * 1024 + t] : -1;
    __syncthreads();
    for (int k = 2; k <= 1024; k <<= 1) {
        for (int j = k >> 1; j > 0; j >>= 1) {
            int ixj = t ^ j;
            if (ixj > t) {
                bool up = ((t & k) == 0);
                float a = key[t], b = key[ixj];
                bool sw = up ? (a > b) : (a < b);
                if (sw) {
                    key[t] = b; key[ixj] = a;
                    int tp = pay[t]; pay[t] = pay[ixj]; pay[ixj] = tp;
                }
            }
            __syncthreads();
        }
    }
    selScore[img * 1024 + t] = -key[t];
    selIdx[img * 1024 + t] = pay[t];
}

// ---------------- K6: box decode + clip + max_coord + class-offset boxes ----------------
__global__ void __launch_bounds__(1024) k_boxes(const float* __restrict__ r0, const float* __restrict__ r1,
                                                const float* __restrict__ r2, const float* __restrict__ r3,
                                                const float* __restrict__ r4,
                                                const int* __restrict__ image_sizes,
                                                const int* __restrict__ candRow, const int* __restrict__ candLvl,
                                                const float* __restrict__ selScore, const int* __restrict__ selIdx,
                                                float* __restrict__ selBox, float* __restrict__ nbBox,
                                                int* __restrict__ selLab) {
    __shared__ float red[1024];
    int t = threadIdx.x, img = blockIdx.x;
    float h = (float)image_sizes[img * 2 + 0];
    float w = (float)image_sizes[img * 2 + 1];
    float bx0 = 0.f, bx1 = 0.f, bx2 = 0.f, bx3 = 0.f;
    int lab = 0; bool valid = false;
    if (t < MSEL) {
        int flat = selIdx[img * 1024 + t];
        if (flat < 0) flat = 0;
        int cand = flat / NCLS;
        lab = flat - cand * NCLS + 1;
        int lvl = candLvl[img * 3280 + cand];
        int row = candRow[img * 3280 + cand];
        const float* rp; int hw; float s;
        if (lvl == 0)      { rp = r0; hw = 15200; s = 8.f; }
        else if (lvl == 1) { rp = r1; hw = 3800;  s = 16.f; }
        else if (lvl == 2) { rp = r2; hw = 950;   s = 32.f; }
        else if (lvl == 3) { rp = r3; hw = 247;   s = 64.f; }
        else               { rp = r4; hw = 70;    s = 128.f; }
        const float* pb = rp + (size_t)img * 4 * hw + row;
        float v0 = pb[0], v1 = pb[(size_t)hw], v2 = pb[(size_t)2 * hw], v3 = pb[(size_t)3 * hw];
        // min(reg*stride, [h,w]) pairs, then reorder [1,0,3,2]
        float x1 = fminf(v1 * s, w), y1 = fminf(v0 * s, h);
        float x2 = fminf(v3 * s, w), y2 = fminf(v2 * s, h);
        x2 = fmaxf(x1 + 1.f, x2); y2 = fmaxf(y1 + 1.f, y2);
        x1 = fminf(fmaxf(x1, 0.f), w - 1.f); y1 = fminf(fmaxf(y1, 0.f), h - 1.f);
        x2 = fminf(fmaxf(x2, 0.f), w - 1.f); y2 = fminf(fmaxf(y2, 0.f), h - 1.f);
        bx0 = x1; bx1 = y1; bx2 = x2; bx3 = y2;
        valid = selScore[img * 1024 + t] > THR_LOGIT;
    }
    red[t] = valid ? fmaxf(fmaxf(bx0, bx1), fmaxf(bx2, bx3)) : 0.f;
    __syncthreads();
    for (int s2 = 512; s2 > 0; s2 >>= 1) {
        if (t < s2) red[t] = fmaxf(red[t], red[t + s2]);
        __syncthreads();
    }
    float off = (float)lab * (red[0] + 1.f);
    size_t o = (size_t)img * 4096 + (size_t)t * 4;
    selBox[o + 0] = bx0; selBox[o + 1] = bx1; selBox[o + 2] = bx2; selBox[o + 3] = bx3;
    nbBox[o + 0] = bx0 + off; nbBox[o + 1] = bx1 + off;
    nbBox[o + 2] = bx2 + off; nbBox[o + 3] = bx3 + off;
    selLab[img * 1024 + t] = lab;
}

// ---------------- K7: suppression bitmask via WMMA area-sum + TDM box staging -------
// One wave per block; block = (img, 16-row I-tile). Loops 64 J-tiles of 16 cols.
__global__ void __launch_bounds__(32) k_mask(const float* __restrict__ nbBox,
                                             unsigned* __restrict__ mask) {
    __shared__ float lb[1024 * 4];           // 16 KB of offset boxes, TDM-staged
    int img = blockIdx.x / 63, itile = blockIdx.x % 63;
    int lane = threadIdx.x;

    // --- Tensor Data Mover: 4096 f32 contiguous global -> LDS ---
    unsigned long long ga = (unsigned long long)(const void*)(nbBox + (size_t)img * 4096);
    unsigned ldsOff = (unsigned)(size_t)(void*)lb;
    u32x4 g0;
    g0.x = 1u;                                              // count=1, user descriptor
    g0.y = ldsOff;                                          // lds_addr
    g0.z = (unsigned)ga;                                    // global_addr[31:0]
    g0.w = (unsigned)((ga >> 32) & 0x01FFFFFFull) | 0x80000000u; // addr[56:32] | type=2
    i32x8 g1;
    g1[0] = 0x00020000;                 // data_size=4B (2<<16), workgroup_mask=0
    g1[1] = (int)((4096u & 0xFFFFu) << 16);   // tensor_dim0 low16 -> bits[31:16]
    g1[2] = 0x00010000;                 // tensor_dim0 hi16=0 | tensor_dim1=1 (low16<<16)
    g1[3] = (int)(4096u << 16);         // tensor_dim1 hi16=0 | tile_dim0=4096
    g1[4] = 0;                          // tile_dim1=0 (unused), tile_dim2=0
    g1[5] = 4096;                       // tensor_dim0_stride[31:0]
    g1[6] = 0;                          // stride hi, dim1_stride lo
    g1[7] = 0;
    i32x4 gz; gz[0] = 0; gz[1] = 0; gz[2] = 0; gz[3] = 0;
    i32x8 gz8; gz8[0] = 0; gz8[1] = 0; gz8[2] = 0; gz8[3] = 0;
    gz8[4] = 0; gz8[5] = 0; gz8[6] = 0; gz8[7] = 0;
    __builtin_amdgcn_tensor_load_to_lds(g0, g1, gz, gz, gz8, 0);
    __builtin_amdgcn_s_wait_tensorcnt(0);

    int I0 = itile * 16;
    int hf = lane >> 4;                 // 0: M=0..7, 1: M=8..15
    int m = lane & 15;

    // preload the 8 I-rows this lane's D components correspond to
    float bi[8][4];
    #pragma unroll
    for (int r = 0; r < 8; ++r) {
        int row = I0 + hf * 8 + r;
        #pragma unroll
        for (int q = 0; q < 4; ++q) bi[r][q] = lb[row * 4 + q];
    }
    // A operand: A[m] = [area(I0+m), 1, 0, 0]   (lanes 0-15 hold K=0,1; lanes 16-31 K=2,3 = 0)
    float aiA;
    {
        const float* p = &lb[(I0 + m) * 4];
        aiA = (p[2] - p[0]) * (p[3] - p[1]);
    }
    v2f A; A.x = (lane < 16) ? aiA : 0.f; A.y = (lane < 16) ? 1.f : 0.f;

    unsigned acc = 0;
    for (int jt = 0; jt < 64; ++jt) {
        int J0 = jt * 16;
        const float* pj = &lb[(J0 + m) * 4];
        float bj0 = pj[0], bj1 = pj[1], bj2 = pj[2], bj3 = pj[3];
        float aj = (bj2 - bj0) * (bj3 - bj1);
        // B operand: B[:,n] = [1, area(J0+n), 0, 0]
        v2f Bv; Bv.x = (lane < 16) ? 1.f : 0.f; Bv.y = (lane < 16) ? aj : 0.f;
        v8f Cz = {0.f, 0.f, 0.f, 0.f, 0.f, 0.f, 0.f, 0.f};
        // D[i][j] = area_i + area_j  via 16x16x4 f32 WMMA outer-sum
        v8f D = __builtin_amdgcn_wmma_f32_16x16x4_f32(false, A, false, Bv,
                                                      (short)0, Cz, false, false);
        unsigned bal[8];
        #pragma unroll
        for (int r = 0; r < 8; ++r) {
            float ix1 = fmaxf(bi[r][0], bj0), iy1 = fmaxf(bi[r][1], bj1);
            float ix2 = fminf(bi[r][2], bj2), iy2 = fminf(bi[r][3], bj3);
            float inter = fmaxf(ix2 - ix1, 0.f) * fmaxf(iy2 - iy1, 0.f);
            float denom = fmaxf(D[r] - inter, 1e-9f);
            bal[r] = __builtin_amdgcn_ballot_w32(inter > NMS_T * denom);
        }
        if (lane < 16) {
            // lane l owns mask row I0+l; its 16 bits are ballot[l&7] half (l>>3)
            int r = lane & 7;
            unsigned bits = (lane >> 3) ? (bal[r] >> 16) : (bal[r] & 0xFFFFu);
            acc |= bits << (16 * (jt & 1));
            if (jt & 1) {
                int row = I0 + lane;
                if (row < MSEL) mask[(size_t)img * 32768 + (size_t)row * 32 + (jt >> 1)] = acc;
                acc = 0;
            }
        }
    }
}

// ---------------- K8: greedy NMS scan, one wave per image ----------------
__global__ void __launch_bounds__(32) k_nms(const float* __restrict__ selScore,
                                            const unsigned* __restrict__ mask,
                                            unsigned* __restrict__ keep) {
    int img = blockIdx.x, lane = threadIdx.x;
    __shared__ float sc[1024];
    for (int i = lane; i < 1024; i += 32) sc[i] = selScore[img * 1024 + i];
    __syncthreads();
    unsigned supp = 0, keepw = 0;
    const unsigned* mrow = mask + (size_t)img * 32768;
    for (int i = 0; i < MSEL; ++i) {
        unsigned sw = (unsigned)__shfl((int)supp, i >> 5, 32);
        bool kept = (sc[i] > THR_LOGIT) && !((sw >> (i & 31)) & 1u);
        if (kept) {
            supp |= mrow[(size_t)i * 32 + lane];
            if (lane == (i >> 5)) keepw |= 1u << (i & 31);
        }
    }
    keep[img * 32 + lane] = keepw;
}

// ---------------- K9: final top-100 assembly ----------------
// Candidates are score-descending; kept-in-order are the top-k of kept_s,
// then slots fill with lowest-index unkept (score NEG) entries, matching jax top_k.
__global__ void __launch_bounds__(32) k_final(const float* __restrict__ selScore,
                                              const int* __restrict__ selLab,
                                              const float* __restrict__ selBox,
                                              const unsigned* __restrict__ keep,
                                              float* __restrict__ out) {
    int img = blockIdx.x;
    if (threadIdx.x != 0) return;
    const unsigned* kw = keep + img * 32;
    float* ob = out + (size_t)img * 400;
    float* os = out + 6400 + (size_t)img * 100;
    float* ol = out + 8000 + (size_t)img * 100;
    float* ov = out + 9600 + (size_t)img * 100;
    int slot = 0;
    for (int pass = 0; pass < 2 && slot < 100; ++pass) {
        for (int i = 0; i < MSEL && slot < 100; ++i) {
            bool k = (kw[i >> 5] >> (i & 31)) & 1u;
            if ((pass == 0) == k) {
                const float* bp = selBox + (size_t)img * 4096 + (size_t)i * 4;
                ob[slot * 4 + 0] = bp[0]; ob[slot * 4 + 1] = bp[1];
                ob[slot * 4 + 2] = bp[2]; ob[slot * 4 + 3] = bp[3];
                float lg = selScore[img * 1024 + i];
                os[slot] = (pass == 0) ? 1.f / (1.f + expf(-lg)) : 0.f;
                ol[slot] = (float)selLab[img * 1024 + i];
                ov[slot] = (pass == 0) ? 1.f : 0.f;
                ++slot;
            }
        }
    }
}

// ---------------- host-side launcher ----------------
extern "C" void kernel_launch(void* const* d_in, const int* in_sizes, int n_in,
                              void* d_out, int out_size, void* d_ws, size_t ws_size,
                              hipStream_t stream) {
    // setup_inputs() dict order is INTERLEAVED: cls0,reg0,cls1,reg1,...,image_sizes
    const float* cls[5] = { (const float*)d_in[0], (const float*)d_in[2], (const float*)d_in[4],
                            (const float*)d_in[6], (const float*)d_in[8] };
    const float* reg[5] = { (const float*)d_in[1], (const float*)d_in[3], (const float*)d_in[5],
                            (const float*)d_in[7], (const float*)d_in[9] };
    const int* image_sizes = (const int*)d_in[10];
    float* out = (float*)d_out;

    char* w = (char*)d_ws;
    float*    rowmax  = (float*)(w + WS_ROWMAX);
    int*      candRow = (int*)(w + WS_CANDROW);
    int*      candLvl = (int*)(w + WS_CANDLVL);
    float*    clog    = (float*)(w + WS_CANDLOG);
    float*    selSc   = (float*)(w + WS_SELSC);
    int*      selIdx  = (int*)(w + WS_SELIDX);
    float*    selBox  = (float*)(w + WS_SELBOX);
    float*    nbBox   = (float*)(w + WS_NBBOX);
    int*      selLab  = (int*)(w + WS_SELLAB);
    unsigned* mask    = (unsigned*)(w + WS_MASK);
    unsigned* keep    = (unsigned*)(w + WS_KEEP);
    int*      cnt     = (int*)(w + WS_CNT);

    k_init<<<1, 256, 0, stream>>>(cnt);
    k_rowmax<<<(NB * 19000 + 255) / 256, 256, 0, stream>>>(cls[0], cls[1], rowmax);
    k_level_select<<<NB * 2, 1024, 0, stream>>>(rowmax, candRow, candLvl, cnt);
    k_fill_tail<<<(NB * 1267 + 255) / 256, 256, 0, stream>>>(candRow, candLvl);
    k_gather<<<(NB * NCAND * NCLS + 255) / 256, 256, 0, stream>>>(
        cls[0], cls[1], cls[2], cls[3], cls[4], candRow, candLvl, clog);
    k_flat_select<<<NB, 1024, 0, stream>>>(clog, selSc, selIdx, cnt + 64);
    k_sort<<<NB, 1024, 0, stream>>>(selSc, selIdx);
    k_boxes<<<NB, 1024, 0, stream>>>(reg[0], reg[1], reg[2], reg[3], reg[4], image_sizes,
                                     candRow, candLvl, selSc, selIdx, selBox, nbBox, selLab);
    k_mask<<<NB * 63, 32, 0, stream>>>(nbBox, mask);
    k_nms<<<NB, 32, 0, stream>>>(selSc, mask, keep);
    k_final<<<NB, 32, 0, stream>>>(selSc, selLab, selBox, keep, out);
}